// GConvGRUModel_49529562857568
// MI455X (gfx1250) — compile-verified
//
#include <hip/hip_runtime.h>
#include <math.h>

typedef __attribute__((ext_vector_type(2))) float v2f;
typedef __attribute__((ext_vector_type(8))) float v8f;
typedef unsigned int __attribute__((ext_vector_type(4))) v4u;
typedef int __attribute__((ext_vector_type(4))) v4i;
typedef int __attribute__((ext_vector_type(8))) v8i;

#define N_NODES        100000
#define F_IN           512
#define H_DIM          32
#define ROWS_PER_BLOCK 128
#define KCHUNK         64
#define NCHUNKS        (F_IN / KCHUNK)       // 8
#define XSTRIDE        68                    // 64 + 4 pad DWORDs (TDM pad_amount=4)
#define XBUF_FLOATS    (ROWS_PER_BLOCK * XSTRIDE)
#define WCHUNK_FLOATS  (16 * 64 * 4)         // 16 q-steps x 64 cols x 4 k-sub = 4096 floats (16KB)
#define WSWZ_FLOATS    (128 * 64 * 4)        // full pre-swizzled weights: 32768 floats (128KB)

// ---- Tensor Data Mover: 2D tile load Global -> LDS (TENSORcnt-tracked) ----
// D# packing per CDNA5 ISA. Pad applied iff codes nonzero.
__device__ __forceinline__ void tdm_load_2d(
    unsigned lds_off, const void* gaddr,
    unsigned tensor_d0, unsigned tensor_d1,      // tensor extents (elements)
    unsigned tile_d0,   unsigned tile_d1,        // tile extents (elements)
    unsigned stride0,                            // row stride in elements
    unsigned pad_interval_code, unsigned pad_amount_code)
{
    unsigned long long ga = (unsigned long long)gaddr;
    v4u g0;
    g0.x = 1u;                                            // count=1 (valid user descriptor)
    g0.y = lds_off;                                       // lds_addr (byte offset)
    g0.z = (unsigned)(ga & 0xFFFFFFFFull);                // global_addr[31:0]
    g0.w = (unsigned)((ga >> 32) & 0x01FFFFFFull)         // global_addr[56:32]
         | (2u << 30);                                    // type = 2 ("image")
    unsigned pad_en = (pad_interval_code | pad_amount_code) ? (1u << 20) : 0u;
    v8i g1;
    g1[0] = (int)((2u << 16)                              // data_size = 2 -> 4 bytes
                | pad_en
                | (pad_interval_code << 22)
                | (pad_amount_code   << 25));
    g1[1] = (int)((tensor_d0 & 0xFFFFu) << 16);           // tensor_dim0[15:0] @ bits63:48
    g1[2] = (int)((tensor_d0 >> 16) | ((tensor_d1 & 0xFFFFu) << 16));
    g1[3] = (int)((tensor_d1 >> 16) | (tile_d0 << 16));   // tile_dim0 @ bits127:112
    g1[4] = (int)tile_d1;                                 // tile_dim1; tile_dim2=0 (2D)
    g1[5] = (int)stride0;                                 // tensor_dim0_stride[31:0]
    g1[6] = 0;
    g1[7] = 0;
    v4i z4 = {0, 0, 0, 0};
    v8i z8 = {0, 0, 0, 0, 0, 0, 0, 0};
    __builtin_amdgcn_tensor_load_to_lds(g0, g1, z4, z4, z8, 0);
}

// ---- Kernel A: pre-swizzle weights into fragment-ready layout in d_ws ----
// wsz[((k/4)*64 + col)*4 + (k%4)] ; cols 0-31 = W_xz, 32-63 = W_xh.
// Per-lane B fragment then = one contiguous float2 -> ds_load_b64, no VGPR shuffles.
__global__ __launch_bounds__(256) void swizzle_weights(
    const float* __restrict__ W_xz, const float* __restrict__ W_xh,
    float* __restrict__ wsz)
{
    int i = blockIdx.x * 256 + threadIdx.x;        // 0 .. 16383
    if (i < F_IN * H_DIM) {
        int k = i >> 5, h = i & 31;
        int q = k >> 2, c = k & 3;
        wsz[(q * 64 + h     ) * 4 + c] = W_xz[i];
        wsz[(q * 64 + h + 32) * 4 + c] = W_xh[i];
    }
}

__global__ __launch_bounds__(256) void gconvgru_fused(
    const float* __restrict__ x,
    const float* __restrict__ wsz,                 // pre-swizzled weights (d_ws)
    const float* __restrict__ b_xz, const float* __restrict__ b_hz,
    const float* __restrict__ b_xh, const float* __restrict__ b_hh,
    const float* __restrict__ W_fc, const float* __restrict__ b_fc,
    float* __restrict__ out)
{
    __shared__ float xs[2][XBUF_FLOATS];    // x chunk, rows padded to stride 68 by TDM
    __shared__ float ws[2][WCHUNK_FLOATS];  // swizzled weight chunk [q16][col64][c4]

    const int t    = threadIdx.x;
    const int wave = t >> 5;
    const int lane = t & 31;
    const int l16  = lane & 15;
    const int hi   = lane >> 4;             // fragment half: K{0,1} vs K{2,3}
    const int blockRow = blockIdx.x * ROWS_PER_BLOCK;

    auto issue_chunk = [&](int kc, int b) {
        unsigned xoff = (unsigned)(unsigned long long)&xs[b][0];
        unsigned woff = (unsigned)(unsigned long long)&ws[b][0];
        // x tile: 64 cols x 128 rows; rows beyond N read as zero (TDM OOB handles tail)
        tdm_load_2d(xoff, x + (size_t)blockRow * F_IN + kc * KCHUNK,
                    F_IN, (unsigned)(N_NODES - blockRow),
                    KCHUNK, ROWS_PER_BLOCK, F_IN,
                    5 /*pad every 64 DW*/, 3 /*pad 4 DW -> stride 68*/);
        // weight slab: contiguous 4096 floats, expressed as 16 rows x 256 (no pad)
        tdm_load_2d(woff, wsz + (size_t)kc * WCHUNK_FLOATS,
                    256, 16, 256, 16, 256, 0, 0);
    };

    v8f acc0 = {}, acc1 = {}, acc2 = {}, acc3 = {};

    if (t < 32) issue_chunk(0, 0);          // prologue DMA

    for (int kc = 0; kc < NCHUNKS; ++kc) {
        const int buf = kc & 1;
        __builtin_amdgcn_s_wait_tensorcnt(0);   // wave0: chunk kc landed (others: no-op)
        __syncthreads();                        // LDS visible; other buffer free
        if (t < 32 && kc + 1 < NCHUNKS)
            issue_chunk(kc + 1, buf ^ 1);       // DMA for next chunk overlaps WMMAs

        const float* xb = &xs[buf][(wave * 16 + l16) * XSTRIDE + hi * 2];
        const float* wb = &ws[buf][l16 * 4 + hi * 2];
        #pragma unroll
        for (int q = 0; q < 16; ++q) {
            v2f a  = *reinterpret_cast<const v2f*>(xb + q * 4);          // A: K=4q+2hi,+1
            v2f b0 = *reinterpret_cast<const v2f*>(wb + q * 256 +   0);  // cols  0-15
            v2f b1 = *reinterpret_cast<const v2f*>(wb + q * 256 +  64);  // cols 16-31
            v2f b2 = *reinterpret_cast<const v2f*>(wb + q * 256 + 128);  // cols 32-47
            v2f b3 = *reinterpret_cast<const v2f*>(wb + q * 256 + 192);  // cols 48-63
            acc0 = __builtin_amdgcn_wmma_f32_16x16x4_f32(false, a, false, b0, (short)0, acc0, false, false);
            acc1 = __builtin_amdgcn_wmma_f32_16x16x4_f32(false, a, false, b1, (short)0, acc1, false, false);
            acc2 = __builtin_amdgcn_wmma_f32_16x16x4_f32(false, a, false, b2, (short)0, acc2, false, false);
            acc3 = __builtin_amdgcn_wmma_f32_16x16x4_f32(false, a, false, b3, (short)0, acc3, false, false);
        }
        __syncthreads();                        // done reading buf before refill
    }

    // ---- fused GRU epilogue + fc ----
    // C/D layout: lane L, VGPR r -> row = r + 8*(L>=16), col = L%16 (per 16-col tile)
    float bz0 = b_xz[l16]      + b_hz[l16];
    float bz1 = b_xz[16 + l16] + b_hz[16 + l16];
    float bh0 = b_xh[l16]      + b_hh[l16];
    float bh1 = b_xh[16 + l16] + b_hh[16 + l16];
    float wf0 = W_fc[l16];
    float wf1 = W_fc[16 + l16];
    float bfc = b_fc[0];

    #pragma unroll
    for (int r = 0; r < 8; ++r) {
        // 1 - sigmoid(a) = 1/(1+e^a)
        float omz0 = 1.0f / (1.0f + __expf(acc0[r] + bz0));
        float omz1 = 1.0f / (1.0f + __expf(acc1[r] + bz1));
        float g0 = omz0 * tanhf(acc2[r] + bh0);
        float g1 = omz1 * tanhf(acc3[r] + bh1);
        float p  = fmaxf(g0, 0.0f) * wf0 + fmaxf(g1, 0.0f) * wf1;
        #pragma unroll
        for (int m = 8; m >= 1; m >>= 1)        // 16-lane half-wave reduction
            p += __shfl_xor(p, m, 32);
        int row = blockRow + wave * 16 + r + hi * 8;
        if (l16 == 0 && row < N_NODES)
            out[row] = p + bfc;
    }
}

extern "C" void kernel_launch(void* const* d_in, const int* in_sizes, int n_in,
                              void* d_out, int out_size, void* d_ws, size_t ws_size,
                              hipStream_t stream) {
    (void)in_sizes; (void)n_in; (void)out_size; (void)ws_size;
    // 0:x 1:edge_index 2:edge_weight 3:W_xz 4:b_xz 5:W_hz 6:b_hz 7:W_xr 8:b_xr
    // 9:W_hr 10:b_hr 11:W_xh 12:b_xh 13:W_hh 14:b_hh 15:W_fc 16:b_fc
    const float* x    = (const float*)d_in[0];
    const float* W_xz = (const float*)d_in[3];
    const float* b_xz = (const float*)d_in[4];
    const float* b_hz = (const float*)d_in[6];
    const float* W_xh = (const float*)d_in[11];
    const float* b_xh = (const float*)d_in[12];
    const float* b_hh = (const float*)d_in[14];
    const float* W_fc = (const float*)d_in[15];
    const float* b_fc = (const float*)d_in[16];
    float* out = (float*)d_out;
    float* wsz = (float*)d_ws;                 // 131072 bytes of scratch used

    swizzle_weights<<<(F_IN * H_DIM + 255) / 256, 256, 0, stream>>>(W_xz, W_xh, wsz);

    int blocks = (N_NODES + ROWS_PER_BLOCK - 1) / ROWS_PER_BLOCK;  // 782
    gconvgru_fused<<<blocks, 256, 0, stream>>>(
        x, wsz, b_xz, b_hz, b_xh, b_hh, W_fc, b_fc, out);
}